// GCN_75007308858124
// MI455X (gfx1250) — compile-verified
//
#include <hip/hip_runtime.h>
#include <hip/hip_bf16.h>
#include <math.h>

// ---------------- problem constants (fixed by the reference) ----------------
#define NTOT   46080      // B*NODES
#define EDGES  737280     // N*DEG
#define GRAPHS 512
#define GNODES 90
#define GEDGES 1440       // edges per graph, contiguous
#define HID    128
#define FIN    90
#define HLIN   64
#define KLIN   34560      // 90*3*128
#define KC     32         // K-chunk staged in LDS per GEMM block
#define GWAVES 8          // waves per GEMM block (M-tiles)

typedef float v2f __attribute__((ext_vector_type(2)));
typedef float v8f __attribute__((ext_vector_type(8)));

// ---------------------------------------------------------------------------
// degree / norm preparation
// ---------------------------------------------------------------------------
__global__ void k_deg_init(float* deg, int n) {
    int i = blockIdx.x * blockDim.x + threadIdx.x;
    if (i < n) deg[i] = 1.0f;                       // self-loop weight
}

__global__ void k_deg_accum(const float* __restrict__ ea,
                            const int* __restrict__ dst,
                            float* __restrict__ deg, int e) {
    int i = blockIdx.x * blockDim.x + threadIdx.x;
    if (i < e) atomicAdd(&deg[dst[i]], ea[i]);
}

__global__ void k_dis(float* __restrict__ deg_to_dis,
                      float* __restrict__ selfnorm, int n) {
    int i = blockIdx.x * blockDim.x + threadIdx.x;
    if (i < n) {
        float r = rsqrtf(deg_to_dis[i]);            // deg >= 1 always
        deg_to_dis[i] = r;
        selfnorm[i] = r * r;
    }
}

__global__ void k_norm(const float* __restrict__ ea,
                       const int* __restrict__ src,
                       const int* __restrict__ dst,
                       const float* __restrict__ dis,
                       float* __restrict__ nrm, int e) {
    int i = blockIdx.x * blockDim.x + threadIdx.x;
    if (i < e) nrm[i] = dis[src[i]] * ea[i] * dis[dst[i]];
}

__global__ void k_zero(float* p, int n) {
    int i = blockIdx.x * blockDim.x + threadIdx.x;
    if (i < n) p[i] = 0.0f;
}

// ---------------------------------------------------------------------------
// WMMA f32 GEMM:  C[M x Nout] = A[M x K](lda) @ B[K x Nout](ldb)
// Block = 8 waves; wave w owns M-tile (blockIdx.x*8+w)*16, all waves share the
// 16-wide N-panel blockIdx.y. B is staged KC rows at a time into LDS (2 KB),
// zero-padded in the tail so the A path needs only an address clamp (branch-
// less). grid.z splits K (kChunk per slice) with global f32 atomic combine.
// Requirements: M % 128 == 0, Nout % 16 == 0, lda/ldb even, kChunk % 4 == 0.
// ---------------------------------------------------------------------------
__global__ __launch_bounds__(256)
void k_wmma_gemm_f32(const float* __restrict__ A, const float* __restrict__ B,
                     float* __restrict__ C, int K,
                     int lda, int ldb, int ldc, int kChunk, int doAtomic) {
    __shared__ float bS[KC * 16];

    const int tid  = threadIdx.x;
    const int lane = tid & 31;
    const int wv   = tid >> 5;                 // 0..7
    const int half = lane >> 4;                // 0/1 -> K sub-pair
    const int l15  = lane & 15;
    const int tm   = (blockIdx.x * GWAVES + wv) * 16;
    const int tn   = blockIdx.y * 16;
    const int kBegin = blockIdx.z * kChunk;
    int kEnd = kBegin + kChunk; if (kEnd > K) kEnd = K;
    const int aClamp = kEnd - 2;               // even (all call sites)

    v8f c = {0.f, 0.f, 0.f, 0.f, 0.f, 0.f, 0.f, 0.f};
    const float* arow = A + (size_t)(tm + l15) * lda;

    for (int kk = kBegin; kk < kEnd; kk += KC) {
        const int kc = kEnd - kk;              // rows valid in this chunk
        // ---- stage B chunk [KC x 16] into LDS (branchless zero-pad) ----
        for (int idx = tid; idx < KC * 16; idx += 256) {
            const int kr = idx >> 4;
            const int n  = idx & 15;
            const int krc = (kr < kc) ? kr : (kc - 1);          // in-bounds
            const float v = B[(size_t)(kk + krc) * ldb + tn + n];
            bS[idx] = (kr < kc) ? v : 0.0f;
        }
        __syncthreads();

        if (kc >= KC) {                        // fast path: fully in-range
            #pragma unroll
            for (int k4 = 0; k4 < KC; k4 += 4) {
                const int k0 = kk + k4 + half * 2;
                const v2f a = *(const v2f*)(arow + k0);
                v2f b;
                b.x = bS[(k4 + half * 2) * 16 + l15];
                b.y = bS[(k4 + half * 2 + 1) * 16 + l15];
                c = __builtin_amdgcn_wmma_f32_16x16x4_f32(
                        false, a, false, b, (short)0, c, false, false);
            }
        } else {                               // tail: clamp A addr; B rows 0
            #pragma unroll
            for (int k4 = 0; k4 < KC; k4 += 4) {
                int k0 = kk + k4 + half * 2;
                if (k0 > aClamp) k0 = aClamp;  // stays 8B aligned (even)
                const v2f a = *(const v2f*)(arow + k0);
                v2f b;
                b.x = bS[(k4 + half * 2) * 16 + l15];
                b.y = bS[(k4 + half * 2 + 1) * 16 + l15];
                c = __builtin_amdgcn_wmma_f32_16x16x4_f32(
                        false, a, false, b, (short)0, c, false, false);
            }
        }
        __syncthreads();
    }

    // D layout: VGPR r, lane l -> element (M = r + 8*half, N = l15)
    float* crow = C + (size_t)(tm + half * 8) * ldc + tn + l15;
    if (doAtomic) {
        #pragma unroll
        for (int r = 0; r < 8; ++r) atomicAdd(crow + (size_t)r * ldc, c[r]);
    } else {
        #pragma unroll
        for (int r = 0; r < 8; ++r) crow[(size_t)r * ldc] = c[r];
    }
}

// ---------------------------------------------------------------------------
// Per-graph GCN aggregation: out[d] = sum_e norm[e]*tmp[src[e]]
//                                    + selfnorm[d]*tmp[d] + bias, then ReLU.
// One block per graph; 90x128 f32 accumulator + edge cache in LDS;
// scatter via LDS float atomics (ds_add_f32). Output written interleaved
// into xc[N,384] at column offset so the final batch reshape is free.
// ---------------------------------------------------------------------------
__global__ __launch_bounds__(256)
void k_aggregate(const float* __restrict__ tmp,     // [N,128] = H_in @ W
                 const float* __restrict__ nrm,     // [E]
                 const float* __restrict__ selfnorm,// [N]
                 const int*   __restrict__ src,     // [E]
                 const int*   __restrict__ dst,     // [E]
                 const float* __restrict__ bias,    // [128]
                 float* __restrict__ outBase) {     // xc + layer*128 (stride 384)
    __shared__ float acc[GNODES * HID];             // 46080 B
    __shared__ float nrmS[GEDGES];                  //  5760 B
    __shared__ short srcS[GEDGES];                  //  2880 B
    __shared__ short dstS[GEDGES];                  //  2880 B

    const int g    = blockIdx.x;
    const int base = g * GNODES;
    const int tid  = threadIdx.x;

    // init accumulator with self-loop contribution (float4)
    for (int q = tid; q < GNODES * 32; q += 256) {
        const int i  = q >> 5;
        const int fc = (q & 31) << 2;
        const float sn = selfnorm[base + i];
        const float4 t = *(const float4*)(tmp + (size_t)(base + i) * HID + fc);
        *(float4*)(acc + i * HID + fc) =
            make_float4(sn * t.x, sn * t.y, sn * t.z, sn * t.w);
    }
    // cache this graph's edges
    for (int e = tid; e < GEDGES; e += 256) {
        const int ge = g * GEDGES + e;
        nrmS[e] = nrm[ge];
        srcS[e] = (short)(src[ge] - base);
        dstS[e] = (short)(dst[ge] - base);
    }
    __syncthreads();

    // scatter: work item = (edge, feature-quad), LDS float atomics
    for (int w = tid; w < GEDGES * 32; w += 256) {
        const int e  = w >> 5;
        const int fc = (w & 31) << 2;
        const float n = nrmS[e];
        const float4 s = *(const float4*)(tmp + (size_t)(base + srcS[e]) * HID + fc);
        float* arow = acc + dstS[e] * HID + fc;
        atomicAdd(arow + 0, n * s.x);
        atomicAdd(arow + 1, n * s.y);
        atomicAdd(arow + 2, n * s.z);
        atomicAdd(arow + 3, n * s.w);
    }
    __syncthreads();

    // bias + ReLU, write interleaved into xc (row stride 384), float4
    for (int q = tid; q < GNODES * 32; q += 256) {
        const int i  = q >> 5;
        const int fc = (q & 31) << 2;
        const float4 a = *(const float4*)(acc + i * HID + fc);
        const float4 b = *(const float4*)(bias + fc);
        *(float4*)(outBase + (size_t)(base + i) * 384 + fc) =
            make_float4(fmaxf(a.x + b.x, 0.f), fmaxf(a.y + b.y, 0.f),
                        fmaxf(a.z + b.z, 0.f), fmaxf(a.w + b.w, 0.f));
    }
}

// zh = relu(zh + lin1_b)
__global__ void k_bias_relu(float* __restrict__ zh,
                            const float* __restrict__ b, int n) {
    int i = blockIdx.x * blockDim.x + threadIdx.x;
    if (i < n) zh[i] = fmaxf(zh[i] + b[i & (HLIN - 1)], 0.0f);
}

// logits = zh @ lin2_W + lin2_b ; out = log_softmax(logits)
__global__ void k_head(const float* __restrict__ zh,
                       const float* __restrict__ W,   // [64,2]
                       const float* __restrict__ b,   // [2]
                       float* __restrict__ out) {     // [512,2]
    int g = blockIdx.x * blockDim.x + threadIdx.x;
    if (g >= GRAPHS) return;
    const float* r = zh + (size_t)g * HLIN;
    float l0 = b[0], l1 = b[1];
    #pragma unroll 8
    for (int k = 0; k < HLIN; ++k) {
        const float v = r[k];
        l0 += v * W[k * 2 + 0];
        l1 += v * W[k * 2 + 1];
    }
    const float m   = fmaxf(l0, l1);
    const float lse = m + logf(expf(l0 - m) + expf(l1 - m));
    out[g * 2 + 0] = l0 - lse;
    out[g * 2 + 1] = l1 - lse;
}

// ---------------------------------------------------------------------------
extern "C" void kernel_launch(void* const* d_in, const int* in_sizes, int n_in,
                              void* d_out, int out_size, void* d_ws, size_t ws_size,
                              hipStream_t stream) {
    const float* x       = (const float*)d_in[0];   // [N,90]
    const float* ea      = (const float*)d_in[1];   // [E]
    const float* W1      = (const float*)d_in[2];
    const float* b1      = (const float*)d_in[3];
    const float* W2      = (const float*)d_in[4];
    const float* b2      = (const float*)d_in[5];
    const float* W3      = (const float*)d_in[6];
    const float* b3      = (const float*)d_in[7];
    const float* lin1_W  = (const float*)d_in[8];   // [34560,64]
    const float* lin1_b  = (const float*)d_in[9];
    const float* lin2_W  = (const float*)d_in[10];  // [64,2]
    const float* lin2_b  = (const float*)d_in[11];
    const int*   ei      = (const int*)d_in[12];    // [2,E]
    const int*   src     = ei;
    const int*   dst     = ei + EDGES;
    float*       out     = (float*)d_out;

    // workspace layout (floats)
    float* ws       = (float*)d_ws;
    float* dis      = ws;                       // N (deg, then rsqrt in place)
    float* selfnorm = dis + NTOT;               // N
    float* nrm      = selfnorm + NTOT;          // E
    float* tmp      = nrm + EDGES;              // N*128
    float* xc       = tmp + (size_t)NTOT * HID; // N*384
    float* zh       = xc + (size_t)NTOT * 384;  // 512*64

    // 1) normalization coefficients
    k_deg_init <<<(NTOT + 255) / 256, 256, 0, stream>>>(dis, NTOT);
    k_deg_accum<<<(EDGES + 255) / 256, 256, 0, stream>>>(ea, dst, dis, EDGES);
    k_dis      <<<(NTOT + 255) / 256, 256, 0, stream>>>(dis, selfnorm, NTOT);
    k_norm     <<<(EDGES + 255) / 256, 256, 0, stream>>>(ea, src, dst, dis, nrm, EDGES);

    // 2) conv1: tmp = x @ W1 (K=90, tail-handled) ; aggregate -> xc[:,0:128]
    k_wmma_gemm_f32<<<dim3(NTOT / 128, HID / 16, 1), 256, 0, stream>>>(
        x, W1, tmp, FIN, FIN, HID, HID, /*kChunk=*/96, 0);
    k_aggregate<<<GRAPHS, 256, 0, stream>>>(tmp, nrm, selfnorm, src, dst, b1, xc);

    // 3) conv2: tmp = h1 @ W2 ; aggregate -> xc[:,128:256]
    k_wmma_gemm_f32<<<dim3(NTOT / 128, HID / 16, 1), 256, 0, stream>>>(
        xc, W2, tmp, HID, 384, HID, HID, /*kChunk=*/HID, 0);
    k_aggregate<<<GRAPHS, 256, 0, stream>>>(tmp, nrm, selfnorm, src, dst, b2, xc + HID);

    // 4) conv3: tmp = h2 @ W3 ; aggregate -> xc[:,256:384]
    k_wmma_gemm_f32<<<dim3(NTOT / 128, HID / 16, 1), 256, 0, stream>>>(
        xc + HID, W3, tmp, HID, 384, HID, HID, /*kChunk=*/HID, 0);
    k_aggregate<<<GRAPHS, 256, 0, stream>>>(tmp, nrm, selfnorm, src, dst, b3, xc + 2 * HID);

    // 5) lin1: zh = z @ lin1_W (z == xc viewed as [512,34560]); K split 36x960
    k_zero<<<(GRAPHS * HLIN + 255) / 256, 256, 0, stream>>>(zh, GRAPHS * HLIN);
    k_wmma_gemm_f32<<<dim3(GRAPHS / 128, HLIN / 16, 36), 256, 0, stream>>>(
        xc, lin1_W, zh, KLIN, KLIN, HLIN, HLIN, /*kChunk=*/960, 1);
    k_bias_relu<<<(GRAPHS * HLIN + 255) / 256, 256, 0, stream>>>(zh, lin1_b, GRAPHS * HLIN);

    // 6) lin2 + log_softmax
    k_head<<<(GRAPHS + 255) / 256, 256, 0, stream>>>(zh, lin2_W, lin2_b, out);
}